// GAT_Layer_28252294873767
// MI455X (gfx1250) — compile-verified
//
#include <hip/hip_runtime.h>

typedef __attribute__((ext_vector_type(2))) float v2f;
typedef __attribute__((ext_vector_type(8))) float v8f;

#define IN_F 128
#define OUT_F 64
#define NEG_SLOPE 0.2f

// ---------------- order-preserving float <-> uint encoding for atomicMax ----
__device__ __forceinline__ unsigned enc_f(float x) {
    unsigned b = __float_as_uint(x);
    return (b & 0x80000000u) ? ~b : (b | 0x80000000u);
}
__device__ __forceinline__ float dec_f(unsigned e) {
    unsigned b = (e & 0x80000000u) ? (e ^ 0x80000000u) : ~e;
    return __uint_as_float(b);
}

// ---------------- Kernel 1: [xl | xr] = x @ [Wl;Wr]^T + [bl;br] via WMMA ----
// Block = 256 threads (8 waves). Block computes rows [16*bx, 16*bx+16) of all
// 128 output columns (cols 0..63 -> xl, 64..127 -> xr). Wave w owns the 16x16
// tile at col_base = 16*w, accumulated with v_wmma_f32_16x16x4_f32 over K=128.
__global__ __launch_bounds__(256) void gat_gemm_wmma(
    const float* __restrict__ x,
    const float* __restrict__ Wl, const float* __restrict__ bl,
    const float* __restrict__ Wr, const float* __restrict__ br,
    float* __restrict__ xl, float* __restrict__ xr, int N)
{
    __shared__ float xs[16][IN_F + 4];   // +4 pad: spread LDS banks
    const int row0 = blockIdx.x * 16;
    const int t = threadIdx.x;

    // cooperative load of the 16x128 X tile (8 elements per thread)
#pragma unroll
    for (int k = 0; k < 8; ++k) {
        int idx = t + k * 256;
        int r = idx >> 7;          // /128
        int c = idx & 127;
        int gr = row0 + r;
        xs[r][c] = (gr < N) ? x[(size_t)gr * IN_F + c] : 0.0f;
    }
    __syncthreads();

    const int wave = t >> 5;
    const int lane = t & 31;
    const int col_base = wave * 16;                 // 0..112, never straddles 64
    const int n = lane & 15;                        // A: M index, B: N index
    const int koff = (lane < 16) ? 0 : 2;           // ISA A/B VGPR layout split

    const bool left = (col_base < OUT_F);
    const float* __restrict__ Wp = left ? (Wl + (size_t)col_base * IN_F)
                                        : (Wr + (size_t)(col_base - OUT_F) * IN_F);
    const float* __restrict__ bp = left ? bl : br;
    float* __restrict__ Yp = left ? xl : xr;
    const int colmod = col_base & (OUT_F - 1);

    v8f acc = {0.f, 0.f, 0.f, 0.f, 0.f, 0.f, 0.f, 0.f};
#pragma unroll 8
    for (int k0 = 0; k0 < IN_F; k0 += 4) {
        // A fragment (16x4): lane L<16 holds X[M=L][k0 +0,+1]; L>=16 -> +2,+3
        v2f a = *(const v2f*)&xs[n][k0 + koff];
        // B fragment (4x16): B[k][n] = W[col_base+n][k] (same lane split)
        v2f b = *(const v2f*)&Wp[(size_t)n * IN_F + k0 + koff];
        acc = __builtin_amdgcn_wmma_f32_16x16x4_f32(
            false, a, false, b, (short)0, acc, false, false);
    }

    const float bias = bp[colmod + n];
    const int row_off = (lane < 16) ? 0 : 8;        // D layout: VGPR r -> M=r / r+8
#pragma unroll
    for (int r = 0; r < 8; ++r) {
        int orow = row0 + r + row_off;
        if (orow < N)
            Yp[(size_t)orow * OUT_F + colmod + n] = acc[r] + bias;
    }
}

// ---------------- Kernel 2: edge logits + segment max ----------------------
__global__ __launch_bounds__(256) void gat_edge_logits(
    const long long* __restrict__ ei,
    const float* __restrict__ xl, const float* __restrict__ xr,
    const float* __restrict__ att,
    float* __restrict__ e_ws, unsigned* __restrict__ m_enc,
    long long E, long long EN)
{
    long long i = (long long)blockIdx.x * blockDim.x + threadIdx.x;
    if (i >= EN) return;
    long long src, dst;
    if (i < E) { src = ei[i]; dst = ei[E + i]; }
    else       { src = i - E; dst = src; }          // self loop

    const float4* __restrict__ pa = (const float4*)(xl + src * OUT_F);
    const float4* __restrict__ pb = (const float4*)(xr + dst * OUT_F);
    const float4* __restrict__ pt = (const float4*)att;
    float e = 0.f;
#pragma unroll
    for (int q = 0; q < OUT_F / 4; ++q) {
        float4 a = pa[q], b = pb[q], w = pt[q];
        float hx = a.x + b.x; hx = (hx > 0.f) ? hx : NEG_SLOPE * hx;
        float hy = a.y + b.y; hy = (hy > 0.f) ? hy : NEG_SLOPE * hy;
        float hz = a.z + b.z; hz = (hz > 0.f) ? hz : NEG_SLOPE * hz;
        float hw = a.w + b.w; hw = (hw > 0.f) ? hw : NEG_SLOPE * hw;
        e += hx * w.x + hy * w.y + hz * w.z + hw * w.w;
    }
    e_ws[i] = e;
    atomicMax(&m_enc[dst], enc_f(e));
}

// ---------------- Kernel 3: exp, segment sum, weighted accumulate ----------
// One wave32 per edge: 2 channels per lane -> coalesced reads + L2 atomics.
__global__ __launch_bounds__(256) void gat_edge_accum(
    const long long* __restrict__ ei,
    const float* __restrict__ xl, const float* __restrict__ e_ws,
    const unsigned* __restrict__ m_enc,
    float* __restrict__ s_ws, float* __restrict__ acc,
    long long E, long long EN)
{
    const int lane = threadIdx.x & 31;
    long long i = (long long)blockIdx.x * (blockDim.x >> 5) + (threadIdx.x >> 5);
    if (i >= EN) return;
    long long src, dst;
    if (i < E) { src = ei[i]; dst = ei[E + i]; }
    else       { src = i - E; dst = src; }

    float m  = dec_f(m_enc[dst]);
    float ex = __expf(e_ws[i] - m);
    if (lane == 0) atomicAdd(&s_ws[dst], ex);

    const float2 v = *(const float2*)(xl + src * OUT_F + lane * 2);
    float* ap = acc + dst * OUT_F + lane * 2;
    atomicAdd(ap,     ex * v.x);
    atomicAdd(ap + 1, ex * v.y);
}

// ---------------- Kernel 4: normalize + bias + ReLU ------------------------
__global__ __launch_bounds__(256) void gat_finalize(
    const float* __restrict__ acc, const float* __restrict__ s_ws,
    const float* __restrict__ bias, float* __restrict__ out, long long total)
{
    long long i = (long long)blockIdx.x * blockDim.x + threadIdx.x;
    if (i >= total) return;
    long long nid = i >> 6;
    int c = (int)(i & 63);
    float v = acc[i] / s_ws[nid] + bias[c];
    out[i] = (v > 0.f) ? v : 0.f;
}

// ---------------- host launcher --------------------------------------------
extern "C" void kernel_launch(void* const* d_in, const int* in_sizes, int n_in,
                              void* d_out, int out_size, void* d_ws, size_t ws_size,
                              hipStream_t stream)
{
    const float*     x    = (const float*)d_in[0];
    const float*     Wl   = (const float*)d_in[1];
    const float*     bl   = (const float*)d_in[2];
    const float*     Wr   = (const float*)d_in[3];
    const float*     br   = (const float*)d_in[4];
    const float*     att  = (const float*)d_in[5];
    const float*     bias = (const float*)d_in[6];
    const long long* ei   = (const long long*)d_in[7];

    const long long N  = in_sizes[0] / IN_F;
    const long long E  = in_sizes[7] / 2;
    const long long EN = E + N;

    // workspace carve: xl, xr, [acc, s, m_enc] (zeroed together), e
    char* ws = (char*)d_ws;
    float*    xl    = (float*)ws;    ws += (size_t)N * OUT_F * sizeof(float);
    float*    xr    = (float*)ws;    ws += (size_t)N * OUT_F * sizeof(float);
    float*    acc   = (float*)ws;    ws += (size_t)N * OUT_F * sizeof(float);
    float*    s_ws  = (float*)ws;    ws += (size_t)N * sizeof(float);
    unsigned* m_enc = (unsigned*)ws; ws += (size_t)N * sizeof(unsigned);
    float*    e_ws  = (float*)ws;    ws += (size_t)EN * sizeof(float);

    // zero acc + s + m_enc in one contiguous memset (graph-capturable)
    hipMemsetAsync(acc, 0, (size_t)(N * OUT_F + 2 * N) * sizeof(float), stream);

    gat_gemm_wmma<<<dim3((unsigned)((N + 15) / 16)), dim3(256), 0, stream>>>(
        x, Wl, bl, Wr, br, xl, xr, (int)N);

    gat_edge_logits<<<dim3((unsigned)((EN + 255) / 256)), dim3(256), 0, stream>>>(
        ei, xl, xr, att, e_ws, m_enc, E, EN);

    gat_edge_accum<<<dim3((unsigned)((EN + 7) / 8)), dim3(256), 0, stream>>>(
        ei, xl, e_ws, m_enc, s_ws, acc, E, EN);

    gat_finalize<<<dim3((unsigned)((N * OUT_F + 255) / 256)), dim3(256), 0, stream>>>(
        acc, s_ws, bias, (float*)d_out, N * OUT_F);
}